// NSFLayer_16810501997234
// MI455X (gfx1250) — compile-verified
//
#include <hip/hip_runtime.h>
#include <stdint.h>
#include <math.h>

#define DIMS 256
#define KBINS 5
#define RECF 10                     // floats per (dim,bin) record
#define DIMSTRIDE (KBINS * RECF)    // 50 floats/dim  (≡2 mod 4 → DS bank-conflict-free)
#define TAB_KN (DIMS * DIMSTRIDE)   // 12800: start of knots
#define TABF (TAB_KN + DIMS * 4)    // 13824 floats total (54 KB)
#define ROWS_PB 64
#define CHUNK 8
#define LN2F 0.6931471805599453f

// ---------------------------------------------------------------------------
// Prep kernel: 256 threads, thread t = dim t. Builds pre-factored spline
// tables in d_ws (global), later DMA'd to LDS by the main kernel.
// Record (dim,bin): { -xk/wk, 1/wk, yk, delta, dk+dk1-2d, hk*d, hk*dk,
//                     d^2*dk1, 2d^3, d^2*dk }
// Knots: interior x-knots cum_w[1..4] per dim at TAB_KN + t*4.
// ---------------------------------------------------------------------------
__global__ __launch_bounds__(DIMS) void rqs_prep(const float* __restrict__ w,
                                                 const float* __restrict__ h,
                                                 const float* __restrict__ dd,
                                                 float* __restrict__ tab) {
  const int t = threadIdx.x;
  float wv[KBINS], hv[KBINS];
#pragma unroll
  for (int i = 0; i < KBINS; ++i) {
    wv[i] = w[t * KBINS + i];
    hv[i] = h[t * KBINS + i];
  }
  float mw = wv[0], mh = hv[0];
#pragma unroll
  for (int i = 1; i < KBINS; ++i) {
    mw = fmaxf(mw, wv[i]);
    mh = fmaxf(mh, hv[i]);
  }
  float sw = 0.0f, sh = 0.0f;
#pragma unroll
  for (int i = 0; i < KBINS; ++i) {
    wv[i] = expf(wv[i] - mw); sw += wv[i];
    hv[i] = expf(hv[i] - mh); sh += hv[i];
  }
  const float rw = 6.0f / sw, rh = 6.0f / sh;  // softmax * 2B, B=3
  float cw[KBINS + 1], ch[KBINS + 1], der[KBINS + 1];
  cw[0] = -3.0f; ch[0] = -3.0f;
#pragma unroll
  for (int i = 0; i < KBINS; ++i) {
    wv[i] *= rw; hv[i] *= rh;
    cw[i + 1] = cw[i] + wv[i];
    ch[i + 1] = ch[i] + hv[i];
  }
  cw[KBINS] = 3.0f; ch[KBINS] = 3.0f;
  der[0] = 1.0f; der[KBINS] = 1.0f;
#pragma unroll
  for (int j = 1; j < KBINS; ++j) {
    float x = dd[t * (KBINS - 1) + (j - 1)];
    der[j] = log1pf(expf(x));  // softplus
  }
#pragma unroll
  for (int i = 0; i < KBINS; ++i) {
    const float wk = wv[i], hk = hv[i];
    const float invw = 1.0f / wk;
    const float delta = hk * invw;
    const float xk = cw[i], yk = ch[i];
    const float dk = der[i], dk1 = der[i + 1];
    float* r = tab + t * DIMSTRIDE + i * RECF;
    r[0] = -xk * invw;
    r[1] = invw;
    r[2] = yk;
    r[3] = delta;
    r[4] = dk + dk1 - 2.0f * delta;
    r[5] = hk * delta;
    r[6] = hk * dk;
    r[7] = delta * delta * dk1;
    r[8] = 2.0f * delta * delta * delta;
    r[9] = delta * delta * dk;
  }
#pragma unroll
  for (int j = 0; j < 4; ++j) tab[TAB_KN + t * 4 + j] = cw[j + 1];
}

// ---------------------------------------------------------------------------
// Fast device intrinsics (guarded so the host pass still parses)
// ---------------------------------------------------------------------------
__device__ __forceinline__ float fast_rcp(float x) {
#if defined(__HIP_DEVICE_COMPILE__)
  return __builtin_amdgcn_rcpf(x);   // v_rcp_f32
#else
  return 1.0f / x;
#endif
}
__device__ __forceinline__ float fast_log2(float x) {
#if defined(__HIP_DEVICE_COMPILE__)
  return __builtin_amdgcn_logf(x);   // v_log_f32 (log2)
#else
  return log2f(x);
#endif
}

// Wave32 sum: 4 DPP butterfly adds (VALU pipe, no LDS traffic) + one
// cross-half xor16 via DS shuffle. Result valid in all lanes.
__device__ __forceinline__ float wave32_sum(float x) {
#if defined(__HIP_DEVICE_COMPILE__)
  int v;
  v = __builtin_amdgcn_update_dpp(0, __float_as_int(x), 0xB1, 0xf, 0xf, true);
  x += __int_as_float(v);  // xor 1  (quad_perm [1,0,3,2])
  v = __builtin_amdgcn_update_dpp(0, __float_as_int(x), 0x4E, 0xf, 0xf, true);
  x += __int_as_float(v);  // xor 2  (quad_perm [2,3,0,1])
  v = __builtin_amdgcn_update_dpp(0, __float_as_int(x), 0x141, 0xf, 0xf, true);
  x += __int_as_float(v);  // xor 4  (row_half_mirror)
  v = __builtin_amdgcn_update_dpp(0, __float_as_int(x), 0x140, 0xf, 0xf, true);
  x += __int_as_float(v);  // xor 8  (row_mirror)
  x += __shfl_xor(x, 16, 32);  // xor 16 (cross half)
  return x;
#else
  for (int m = 16; m > 0; m >>= 1) x += __shfl_xor(x, m, 32);
  return x;
#endif
}

// Per-element spline evaluation. ~28 VALU + rcp + log + 5 ds_load_b64.
__device__ __forceinline__ void rqs_eval(float u, const float* __restrict__ tab,
                                         int t, float k1, float k2, float k3,
                                         float k4, float& xo, float& ld2) {
  const float uc = fminf(fmaxf(u, -3.0f), 3.0f);
  const int idx = (int)(uc >= k1) + (int)(uc >= k2) + (int)(uc >= k3) + (int)(uc >= k4);
  const float2* r = (const float2*)(tab + t * DIMSTRIDE + idx * RECF);
  const float2 r01 = r[0], r23 = r[1], r45 = r[2], r67 = r[3], r89 = r[4];
  const float bxk = r01.x, invw = r01.y;
  const float yk = r23.x, delta = r23.y;
  const float s = r45.x, e1 = r45.y;
  const float e2 = r67.x, f1 = r67.y;
  const float f2 = r89.x, f3 = r89.y;

  const float th  = fmaf(uc, invw, bxk);       // theta
  const float t2  = th * th;
  const float t1m = th - t2;                   // theta*(1-theta)
  const float om  = 1.0f - th;
  const float om2 = om * om;
  const float den = fmaf(s, t1m, delta);
  const float inv = fast_rcp(den);
  const float num = fmaf(e2, t1m, e1 * t2);
  const float x_in = fmaf(num, inv, yk);
  float g = fmaf(f2, t1m, f1 * t2);
  g = fmaf(f3, om2, g);
  const float larg = g * inv * inv;            // ld_num / den^2
  const float l2 = fast_log2(larg);            // natural-log scale applied at end

  const bool inside = (fabsf(u) <= 3.0f);      // NaN -> false -> passthrough
  xo  = inside ? x_in : u;
  ld2 = inside ? l2 : 0.0f;
}

#if defined(__HIP_DEVICE_COMPILE__) && defined(__gfx1250__) &&              \
    __has_builtin(__builtin_amdgcn_tensor_load_to_lds) &&                   \
    __has_builtin(__builtin_amdgcn_s_wait_tensorcnt)
#define USE_TDM 1
typedef unsigned int u32x4 __attribute__((ext_vector_type(4)));
typedef int i32x4 __attribute__((ext_vector_type(4)));
typedef int i32x8 __attribute__((ext_vector_type(8)));
#else
#define USE_TDM 0
#endif

// ---------------------------------------------------------------------------
// Main kernel: thread t = dim t; block handles ROWS_PB rows.
// Table staged into LDS via Tensor Data Mover (TENSORcnt).
// logdet: DPP wave reduce + 8-wave LDS combine per 8-row chunk.
// ---------------------------------------------------------------------------
__global__ __launch_bounds__(DIMS) void rqs_main(const float* __restrict__ u,
                                                 const float* __restrict__ tabg,
                                                 float* __restrict__ xout,
                                                 float* __restrict__ ldout,
                                                 int batch) {
  __shared__ float tab[TABF];
  __shared__ float red[8][CHUNK];

#if USE_TDM
  if (threadIdx.x < 32) {  // one wave issues the DMA (EXEC ignored by TDM)
    const uint64_t ga = (uint64_t)(uintptr_t)tabg;
    // generic LDS pointer: low 32 bits are the LDS byte offset
    const uint32_t lds = (uint32_t)(uintptr_t)(&tab[0]);
    u32x4 g0;
    g0[0] = 1u;                                    // count=1, no gather
    g0[1] = lds;                                   // lds_addr
    g0[2] = (uint32_t)ga;                          // global_addr[31:0]
    g0[3] = ((uint32_t)(ga >> 32) & 0x01FFFFFFu)   // global_addr[56:32]
            | 0x80000000u;                         // type=2 ("image")
    const int td0 = TABF;                          // dwords (data_size=4B)
    i32x8 g1;
    g1[0] = 0x00020000;                  // data_size=2 (4B); no pad/iterate/mask
    g1[1] = (td0 & 0xFFFF) << 16;        // tensor_dim0[15:0]
    g1[2] = (td0 >> 16) | (1 << 16);     // tensor_dim0[31:16] | tensor_dim1=1
    g1[3] = (td0 & 0xFFFF) << 16;        // tile_dim0
    g1[4] = 1;                           // tile_dim1=1, tile_dim2=0
    g1[5] = td0;                         // tensor_dim0_stride lo
    g1[6] = 0;                           // stride0 hi | stride1 lo
    g1[7] = 0;                           // stride1 hi
    i32x4 z4; z4[0] = 0; z4[1] = 0; z4[2] = 0; z4[3] = 0;
    i32x8 z8;
#pragma unroll
    for (int i = 0; i < 8; ++i) z8[i] = 0;
    __builtin_amdgcn_tensor_load_to_lds(g0, g1, z4, z4, z8, 0);
    __builtin_amdgcn_s_wait_tensorcnt(0);
  }
#else
  for (int i = threadIdx.x; i < TABF; i += DIMS) tab[i] = tabg[i];
#endif
  __syncthreads();

  const int t = threadIdx.x;
  const int wv = t >> 5;
  const int lane = t & 31;
  const float* kn = tab + TAB_KN + t * 4;
  const float k1 = kn[0], k2 = kn[1], k3 = kn[2], k4 = kn[3];

  const int r0 = blockIdx.x * ROWS_PB;
#pragma unroll 1
  for (int c = 0; c < ROWS_PB / CHUNK; ++c) {
    const int rbase = r0 + c * CHUNK;
    float uu[CHUNK];
#pragma unroll
    for (int j = 0; j < CHUNK; ++j) {
      const int row = rbase + j;
      uu[j] = (row < batch)
                  ? __builtin_nontemporal_load(u + (size_t)row * DIMS + t)
                  : 0.0f;
    }
#pragma unroll
    for (int j = 0; j < CHUNK; ++j) {
      const int row = rbase + j;
      float xo, l2;
      rqs_eval(uu[j], tab, t, k1, k2, k3, k4, xo, l2);
      if (row < batch)
        __builtin_nontemporal_store(xo, xout + (size_t)row * DIMS + t);
      l2 = wave32_sum(l2);
      if (lane == 0) red[wv][j] = l2;
    }
    __syncthreads();
    if (t < CHUNK) {
      const int row = rbase + t;
      if (row < batch) {
        float sacc = 0.0f;
#pragma unroll
        for (int ww = 0; ww < 8; ++ww) sacc += red[ww][t];
        ldout[row] = sacc * LN2F;
      }
    }
    __syncthreads();
  }
}

// ---------------------------------------------------------------------------
extern "C" void kernel_launch(void* const* d_in, const int* in_sizes, int n_in,
                              void* d_out, int out_size, void* d_ws,
                              size_t ws_size, hipStream_t stream) {
  const float* u = (const float*)d_in[0];
  const float* w = (const float*)d_in[1];
  const float* h = (const float*)d_in[2];
  const float* dd = (const float*)d_in[3];
  const int batch = in_sizes[0] / DIMS;  // 131072

  float* tabg = (float*)d_ws;            // TABF floats = 54 KB
  float* xout = (float*)d_out;
  float* ldout = xout + (size_t)batch * DIMS;

  rqs_prep<<<1, DIMS, 0, stream>>>(w, h, dd, tabg);
  const int blocks = (batch + ROWS_PB - 1) / ROWS_PB;
  rqs_main<<<blocks, DIMS, 0, stream>>>(u, tabg, xout, ldout, batch);
}